// AVODWH_CENTER_IN_13683765805716
// MI455X (gfx1250) — compile-verified
//
#include <hip/hip_runtime.h>
#include <stdint.h>

// Problem constants (match setup_inputs)
#define NIMG  16
#define NCLS  15
#define HW    65536        // 256*256
#define TOPK  1000
#define NBINS 4096         // key>>18 buckets (scores in (0,1) => bucket < 4064)
#define CHUNK 4096         // hw positions staged per block per channel
#define CAP   4096         // max compacted candidates per image
#define THRESH 0.05f

// ---------------------------------------------------------------------------
// CDNA5 helpers: raw LDS byte offset + async global->LDS copy (ASYNCcnt path)
// ---------------------------------------------------------------------------
typedef __attribute__((address_space(3))) char lds_char_t;

static __device__ __forceinline__ uint32_t lds_byte_off(const void* p) {
  return (uint32_t)(uintptr_t)(lds_char_t*)p;
}

// Copy 4096 floats (16KB) global -> LDS with 4 x global_load_async_to_lds_b128
// per lane (256 lanes). Tracked by ASYNCcnt (4 instructions per wave).
static __device__ __forceinline__ void async_copy_chunk(float* dst,
                                                        const float* src,
                                                        int t) {
#pragma unroll
  for (int i = 0; i < 4; ++i) {
    const int j = ((i << 8) + t) << 2;  // float index, 16B aligned
    uint32_t loff = lds_byte_off(dst + j);
    unsigned long long ga = (unsigned long long)(uintptr_t)(src + j);
    asm volatile("global_load_async_to_lds_b128 %0, %1, off"
                 :: "v"(loff), "v"(ga) : "memory");
  }
}

static __device__ __forceinline__ float sigmoidf_fast(float x) {
  return 1.0f / (1.0f + __expf(-x));
}

// ---------------------------------------------------------------------------
// Pass 1: streaming sigmoid + per-image histogram of score bit-patterns.
// Double-buffered async-to-LDS pipeline over the 15 class planes.
// ---------------------------------------------------------------------------
__global__ __launch_bounds__(256) void k_hist(const float* __restrict__ cls,
                                              const float* __restrict__ confs,
                                              uint32_t* __restrict__ ghist) {
  __shared__ __align__(16) float s_cf[CHUNK];
  __shared__ __align__(16) float s_cls[2][CHUNK];
  __shared__ uint32_t s_h[NBINS];

  const int t    = threadIdx.x;
  const int n    = blockIdx.y;
  const int base = blockIdx.x * CHUNK;

  for (int i = t; i < NBINS; i += 256) s_h[i] = 0;

  // Prologue: async-load confs chunk + channel 0 chunk.
  async_copy_chunk(s_cf, confs + (size_t)n * HW + base, t);
  const float* cbase = cls + ((size_t)n * NCLS) * HW + base;
  async_copy_chunk(s_cls[0], cbase, t);
  asm volatile("s_wait_asynccnt 0x0" ::: "memory");
  __syncthreads();

  for (int i = t; i < CHUNK; i += 256) s_cf[i] = sigmoidf_fast(s_cf[i]);
  __syncthreads();

  for (int c = 0; c < NCLS; ++c) {
    const int cur = c & 1;
    if (c + 1 < NCLS) {
      // Prefetch next channel, then wait until only its 4 loads are in flight
      // => current channel's LDS data is ready.
      async_copy_chunk(s_cls[cur ^ 1], cbase + (size_t)(c + 1) * HW, t);
      asm volatile("s_wait_asynccnt 0x4" ::: "memory");
    } else {
      asm volatile("s_wait_asynccnt 0x0" ::: "memory");
    }
    __syncthreads();

    for (int i = t; i < CHUNK; i += 256) {
      const float s = sigmoidf_fast(s_cls[cur][i]);
      if (s > THRESH) {
        const uint32_t key = __float_as_uint(s * s_cf[i]);  // in (0,1): bit order == value order
        atomicAdd(&s_h[key >> 18], 1u);
      }
    }
    __syncthreads();  // done reading this buffer before it is overwritten
  }

  uint32_t* gh = ghist + (size_t)n * NBINS;
  for (int i = t; i < NBINS; i += 256) {
    const uint32_t v = s_h[i];
    if (v) atomicAdd(&gh[i], v);
  }
}

// ---------------------------------------------------------------------------
// Pass 2: find rank-TOPK threshold bucket per image (tiny).
// ---------------------------------------------------------------------------
__global__ void k_thresh(const uint32_t* __restrict__ ghist,
                         uint32_t* __restrict__ B) {
  const int n = threadIdx.x;
  if (n >= NIMG) return;
  const uint32_t* h = ghist + (size_t)n * NBINS;
  uint32_t cum = 0, b = 0;
  for (int i = NBINS - 1; i >= 0; --i) {
    cum += h[i];
    if (cum >= TOPK) { b = (uint32_t)i; break; }
  }
  B[n] = b;
}

// ---------------------------------------------------------------------------
// Pass 3: recompute scores, compact candidates in the threshold bucket or
// above as (key<<32)|flat_idx packed u64 (flat_idx = hw*NCLS + c < 2^20).
// ---------------------------------------------------------------------------
__global__ __launch_bounds__(256) void k_compact(const float* __restrict__ cls,
                                                 const float* __restrict__ confs,
                                                 const uint32_t* __restrict__ B,
                                                 uint32_t* __restrict__ cnt,
                                                 unsigned long long* __restrict__ cand) {
  const int n = blockIdx.y;
  const uint32_t b = B[n];
  const int hw = blockIdx.x * 256 + threadIdx.x;
  const float cf = sigmoidf_fast(confs[(size_t)n * HW + hw]);
  const float* cbase = cls + ((size_t)n * NCLS) * HW + hw;
  unsigned long long* cn = cand + (size_t)n * CAP;

  for (int c = 0; c < NCLS; ++c) {
    const float s = sigmoidf_fast(cbase[(size_t)c * HW]);
    if (s > THRESH) {
      const uint32_t key = __float_as_uint(s * cf);
      if ((key >> 18) >= b) {
        const uint32_t pos = atomicAdd(&cnt[n], 1u);
        if (pos < CAP)
          cn[pos] = ((unsigned long long)key << 32) |
                    (uint32_t)(hw * NCLS + c);
      }
    }
  }
}

// ---------------------------------------------------------------------------
// Pass 4: per-image bitonic sort (desc) of CAP packed candidates in LDS,
// then decode the top-K rotated rectangles and write outputs.
// ---------------------------------------------------------------------------
__global__ __launch_bounds__(1024) void k_sort_decode(
    const uint32_t* __restrict__ cnt,
    const unsigned long long* __restrict__ cand,
    const float* __restrict__ reg, const float* __restrict__ ctr,
    const float* __restrict__ loc,
    int* __restrict__ o_cls, float* __restrict__ o_score,
    float* __restrict__ o_rot, float* __restrict__ o_valid) {
  __shared__ unsigned long long sm[CAP];
  const int n = blockIdx.x;
  const int t = threadIdx.x;
  uint32_t M = cnt[n];
  if (M > CAP) M = CAP;

  for (int i = t; i < CAP; i += 1024)
    sm[i] = (i < (int)M) ? cand[(size_t)n * CAP + i] : 0ull;
  __syncthreads();

  // Bitonic sort, descending (zeros/invalid sink to the end).
  for (int k = 2; k <= CAP; k <<= 1) {
    for (int j = k >> 1; j > 0; j >>= 1) {
      for (int i = t; i < CAP; i += 1024) {
        const int ix = i ^ j;
        if (ix > i) {
          const unsigned long long a = sm[i], bb = sm[ix];
          const bool desc = (i & k) == 0;
          if (desc ? (a < bb) : (a > bb)) { sm[i] = bb; sm[ix] = a; }
        }
      }
      __syncthreads();
    }
  }

  if (t < TOPK) {
    const unsigned long long p = sm[t];
    const int o = n * TOPK + t;
    int   cls_out = 0;
    float score = 0.0f, vld = 0.0f;
    float r0 = 0.0f, r1 = 0.0f, r2 = 0.0f, r3 = 0.0f, r4 = 0.0f;

    if (p != 0ull) {
      const uint32_t key = (uint32_t)(p >> 32);
      const int idx = (int)(uint32_t)p;
      const int hw = idx / NCLS;
      cls_out = idx - hw * NCLS;
      vld = 1.0f;
      score = sqrtf(fmaxf(__uint_as_float(key), 1e-12f));

      const float* rg = reg + ((size_t)n * 4) * HW + hw;
      const float l  = rg[0],      r  = rg[HW];
      const float tt = rg[2 * HW], bm = rg[3 * HW];
      const float* cc = ctr + ((size_t)n * 2) * HW + hw;
      const float ox = cc[0], oy = cc[HW];
      const float lx = loc[2 * hw], ly = loc[2 * hw + 1];

      const float w = l + r, h = tt + bm;
      const float cx = lx + ox, cy = ly + oy;
      const float x1 = cx - 0.5f * w, y1 = cy - 0.5f * h;
      const float x2 = cx + 0.5f * w, y2 = cy + 0.5f * h;

      // Inclined quad inscribed in the hbb.
      const float px0 = x1 + l,  py0 = y1;
      const float px1 = x2,      py1 = y1 + tt;
      const float px2 = x2 - l,  py2 = y2;
      const float px3 = x1,      py3 = y2 - tt;

      const float mx = 0.25f * (px0 + px1 + px2 + px3);
      const float my = 0.25f * (py0 + py1 + py2 + py3);
      const float ang = atan2f(py1 - py0, px1 - px0);
      const float ca = __cosf(ang), sa = __sinf(ang);

      const float ax0 = px0 - mx, ay0 = py0 - my;
      const float ax1 = px1 - mx, ay1 = py1 - my;
      const float ax2 = px2 - mx, ay2 = py2 - my;
      const float ax3 = px3 - mx, ay3 = py3 - my;
      const float xr0 = ax0 * ca + ay0 * sa, yr0 = -ax0 * sa + ay0 * ca;
      const float xr1 = ax1 * ca + ay1 * sa, yr1 = -ax1 * sa + ay1 * ca;
      const float xr2 = ax2 * ca + ay2 * sa, yr2 = -ax2 * sa + ay2 * ca;
      const float xr3 = ax3 * ca + ay3 * sa, yr3 = -ax3 * sa + ay3 * ca;

      const float wq = fmaxf(fmaxf(xr0, xr1), fmaxf(xr2, xr3)) -
                       fminf(fminf(xr0, xr1), fminf(xr2, xr3));
      const float hq = fmaxf(fmaxf(yr0, yr1), fmaxf(yr2, yr3)) -
                       fminf(fminf(yr0, yr1), fminf(yr2, yr3));
      r0 = mx; r1 = my; r2 = wq; r3 = hq; r4 = ang;
    }

    o_cls[o]   = cls_out;
    o_score[o] = score;
    float* rp = o_rot + (size_t)o * 5;
    rp[0] = r0; rp[1] = r1; rp[2] = r2; rp[3] = r3; rp[4] = r4;
    o_valid[o] = vld;
  }
}

// ---------------------------------------------------------------------------
// Launch
// ---------------------------------------------------------------------------
extern "C" void kernel_launch(void* const* d_in, const int* in_sizes, int n_in,
                              void* d_out, int out_size, void* d_ws, size_t ws_size,
                              hipStream_t stream) {
  (void)in_sizes; (void)n_in; (void)out_size; (void)ws_size;
  const float* loc  = (const float*)d_in[0];  // [HW, 2]
  const float* cls  = (const float*)d_in[1];  // [N, C, H, W]
  const float* reg  = (const float*)d_in[2];  // [N, 4, H, W]
  const float* ctr  = (const float*)d_in[3];  // [N, 2, H, W]
  const float* conf = (const float*)d_in[4];  // [N, 1, H, W]

  // Workspace layout
  char* ws = (char*)d_ws;
  uint32_t* ghist = (uint32_t*)ws;                                 // 16*4096*4 = 262144 B
  uint32_t* Bthr  = (uint32_t*)(ws + 262144);                      // 64 B
  uint32_t* cnt   = (uint32_t*)(ws + 262208);                      // 64 B
  unsigned long long* cand = (unsigned long long*)(ws + 262272);   // 16*4096*8 = 524288 B

  hipMemsetAsync(d_ws, 0, 262272, stream);  // zero hist + thresholds + counters

  dim3 g1(HW / CHUNK, NIMG);
  k_hist<<<g1, 256, 0, stream>>>(cls, conf, ghist);
  k_thresh<<<1, 32, 0, stream>>>(ghist, Bthr);
  dim3 g2(HW / 256, NIMG);
  k_compact<<<g2, 256, 0, stream>>>(cls, conf, Bthr, cnt, cand);

  const int NK = NIMG * TOPK;
  int*   o_cls  = (int*)d_out;
  float* fbase  = (float*)d_out;
  k_sort_decode<<<NIMG, 1024, 0, stream>>>(cnt, cand, reg, ctr, loc,
                                           o_cls,            // [N,K]  int32
                                           fbase + NK,       // [N,K]  score
                                           fbase + 2 * NK,   // [N,K,5] rot
                                           fbase + 7 * NK);  // [N,K]  valid
}